// InceptionBlock_24318104830207
// MI455X (gfx1250) — compile-verified
//
#include <hip/hip_runtime.h>

typedef __attribute__((ext_vector_type(2))) float v2f;
typedef __attribute__((ext_vector_type(8))) float v8f;

#define IN_DIM  128
#define OUT_DIM 128

// Guaranteed-native f32 scatter-add: global_atomic_add_f32 (no-return form,
// tracked with STOREcnt; s_endpgm's implicit wait-idle covers completion).
// Bypasses clang's atomicrmw-fadd expansion heuristics (which can fall back
// to a CAS loop depending on fine-grained-memory assumptions).
__device__ __forceinline__ void global_atomic_add_f32(float* p, float v)
{
    asm volatile("global_atomic_add_f32 %0, %1, off"
                 :
                 : "v"(p), "v"(v)
                 : "memory");
}

// ---------------------------------------------------------------------------
// Edge aggregation. One wave per edge; each lane handles 4 consecutive
// columns (float4 = 16B/lane, 512B/wave = full row, coalesced gather).
// Scatter-add with native global_atomic_add_f32.
// ---------------------------------------------------------------------------
__global__ __launch_bounds__(256)
void scatter_edges(const float* __restrict__ h,
                   const long long* __restrict__ ei,   // [2, nEdges] row-major
                   const float* __restrict__ attr,
                   float* __restrict__ out,
                   int nEdges)
{
    const int lane = threadIdx.x;                       // 0..31
    const int edge = blockIdx.x * blockDim.y + threadIdx.y;
    if (edge >= nEdges) return;

    const long long src = ei[edge];
    const long long tgt = ei[(size_t)nEdges + edge];
    const float w = attr[edge];

    const float4* hr = (const float4*)(h + (size_t)src * OUT_DIM);
    const float4 v = hr[lane];

    float* o = out + (size_t)tgt * OUT_DIM + lane * 4;
    global_atomic_add_f32(o + 0, v.x * w);
    global_atomic_add_f32(o + 1, v.y * w);
    global_atomic_add_f32(o + 2, v.z * w);
    global_atomic_add_f32(o + 3, v.w * w);
}

// ---------------------------------------------------------------------------
// Fused triple GEMM via V_WMMA_F32_16X16X4_F32.
// One wave (32 lanes) computes one 16x16 output tile for all three weight
// matrices, reusing the A fragment (rows of x) across the three B operands.
// block = (32, 8): 8 waves cover the full 128 output columns of one row tile.
// ---------------------------------------------------------------------------
__global__ __launch_bounds__(256)
void gemm3_wmma(const float* __restrict__ x,
                const float* __restrict__ Wln, const float* __restrict__ bln,
                const float* __restrict__ W1,
                const float* __restrict__ W2,
                float* __restrict__ out0,   // x @ W_ln + b_ln
                float* __restrict__ h1,     // x @ W1
                float* __restrict__ h2,     // x @ W2
                int nRows)
{
    const int lane    = threadIdx.x;        // 0..31
    const int colTile = threadIdx.y;        // 0..7
    const int m0 = blockIdx.x * 16;
    const int n0 = colTile * 16;
    if (m0 >= nRows) return;                // uniform per block; EXEC stays full

    const int nlo = lane & 15;              // M (for A) / N (for B,C) index
    const int hi  = lane >> 4;              // lane-half selector

    v8f c0 = {}; v8f c1 = {}; v8f c2 = {};

    // A fragment: lane holds row min(m0+nlo, nRows-1); VGPR pair holds
    // K = ka, ka+1 with ka = k0 + 2*hi  (ISA 32-bit A 16x4 layout).
    int arow = m0 + nlo;
    if (arow >= nRows) arow = nRows - 1;    // tail-safe; exact-tile case unaffected
    const float* xrow = x + (size_t)arow * IN_DIM;

    #pragma unroll 4
    for (int k0 = 0; k0 < IN_DIM; k0 += 4) {
        const int ka = k0 + 2 * hi;
        v2f a;
        a.x = xrow[ka];
        a.y = xrow[ka + 1];

        // B fragment: lanes 0-15 hold K = k0,k0+1 ; lanes 16-31 hold K = k0+2,k0+3
        v2f b;
        b.x = Wln[(size_t)ka * OUT_DIM + n0 + nlo];
        b.y = Wln[(size_t)(ka + 1) * OUT_DIM + n0 + nlo];
        c0 = __builtin_amdgcn_wmma_f32_16x16x4_f32(false, a, false, b,
                                                   (short)0, c0, false, false);

        b.x = W1[(size_t)ka * OUT_DIM + n0 + nlo];
        b.y = W1[(size_t)(ka + 1) * OUT_DIM + n0 + nlo];
        c1 = __builtin_amdgcn_wmma_f32_16x16x4_f32(false, a, false, b,
                                                   (short)0, c1, false, false);

        b.x = W2[(size_t)ka * OUT_DIM + n0 + nlo];
        b.y = W2[(size_t)(ka + 1) * OUT_DIM + n0 + nlo];
        c2 = __builtin_amdgcn_wmma_f32_16x16x4_f32(false, a, false, b,
                                                   (short)0, c2, false, false);
    }

    // C/D layout: VGPR v -> M = v + 8*hi, N = n0 + nlo.
    const float bias = bln[n0 + nlo];
    if (m0 + 16 <= nRows) {
        // Fast path: full tile, no per-row guards (uniform branch, EXEC full).
        #pragma unroll
        for (int v = 0; v < 8; ++v) {
            const size_t off = (size_t)(m0 + v + 8 * hi) * OUT_DIM + n0 + nlo;
            out0[off] = c0[v] + bias;
            h1[off]   = c1[v];
            h2[off]   = c2[v];
        }
    } else {
        #pragma unroll
        for (int v = 0; v < 8; ++v) {
            const int row = m0 + v + 8 * hi;
            if (row < nRows) {
                const size_t off = (size_t)row * OUT_DIM + n0 + nlo;
                out0[off] = c0[v] + bias;
                h1[off]   = c1[v];
                h2[off]   = c2[v];
            }
        }
    }
}

// ---------------------------------------------------------------------------
// Initialize scatter targets with broadcast bias (additive identity base for
// segment_sum + b). d_out is poisoned by the harness, so this must run every
// launch.
// ---------------------------------------------------------------------------
__global__ __launch_bounds__(256)
void init_bias(float* __restrict__ out1, float* __restrict__ out2,
               const float* __restrict__ b1, const float* __restrict__ b2,
               size_t total)
{
    size_t i = (size_t)blockIdx.x * blockDim.x + threadIdx.x;
    const size_t stride = (size_t)gridDim.x * blockDim.x;
    for (; i < total; i += stride) {
        const int col = (int)(i & (OUT_DIM - 1));
        out1[i] = b1[col];
        out2[i] = b2[col];
    }
}

// ---------------------------------------------------------------------------
// Host-side launcher.
// Input order: x, edge_index, edge_attr, edge_index2, edge_attr2,
//              W_ln, b_ln, W1, b1, W2, b2
// Output: concat(x0, x1, x2), each [nRows, 128] f32.
// ---------------------------------------------------------------------------
extern "C" void kernel_launch(void* const* d_in, const int* in_sizes, int n_in,
                              void* d_out, int out_size, void* d_ws, size_t ws_size,
                              hipStream_t stream)
{
    const float*     x   = (const float*)d_in[0];
    const long long* ei1 = (const long long*)d_in[1];
    const float*     ea1 = (const float*)d_in[2];
    const long long* ei2 = (const long long*)d_in[3];
    const float*     ea2 = (const float*)d_in[4];
    const float*     Wln = (const float*)d_in[5];
    const float*     bln = (const float*)d_in[6];
    const float*     W1  = (const float*)d_in[7];
    const float*     b1  = (const float*)d_in[8];
    const float*     W2  = (const float*)d_in[9];
    const float*     b2  = (const float*)d_in[10];

    const int nRows = in_sizes[0] / IN_DIM;
    const int nE1   = in_sizes[1] / 2;
    const int nE2   = in_sizes[3] / 2;
    const size_t mat = (size_t)nRows * OUT_DIM;

    float* out0 = (float*)d_out;
    float* out1 = out0 + mat;
    float* out2 = out1 + mat;

    float* h1 = (float*)d_ws;        // [nRows, 128]
    float* h2 = h1 + mat;            // [nRows, 128]

    // 1) fused triple GEMM (WMMA)
    {
        dim3 block(32, 8);
        int rowTiles = (nRows + 15) / 16;
        gemm3_wmma<<<rowTiles, block, 0, stream>>>(x, Wln, bln, W1, W2,
                                                   out0, h1, h2, nRows);
    }

    // 2) bias-broadcast init of scatter targets
    init_bias<<<2048, 256, 0, stream>>>(out1, out2, b1, b2, mat);

    // 3) edge aggregation (depends on h1/h2 and bias init; same stream orders)
    {
        dim3 block(32, 8);  // 8 edges per block
        scatter_edges<<<(nE1 + 7) / 8, block, 0, stream>>>(h1, ei1, ea1, out1, nE1);
        scatter_edges<<<(nE2 + 7) / 8, block, 0, stream>>>(h2, ei2, ea2, out2, nE2);
    }
}